// GCNNBlock_9165460210278
// MI455X (gfx1250) — compile-verified
//
#include <hip/hip_runtime.h>

#define N_NODES 100000
#define N_EDGES 1600000
#define IN_F 32
#define OUT_F 8
#define NTILES (N_NODES / 16)   // 6250, exact

typedef __attribute__((ext_vector_type(2))) float v2f;
typedef __attribute__((ext_vector_type(8))) float v8f;

// ---------------------------------------------------------------------------
// Zero scratch (deg + tx1 + p2). Harness poisons d_ws; re-init every call.
// ---------------------------------------------------------------------------
__global__ void cheb_zero_kernel(float* __restrict__ p, int n) {
    int i = blockIdx.x * blockDim.x + threadIdx.x;
    if (i < n) p[i] = 0.0f;
}

// ---------------------------------------------------------------------------
// deg[s] += 1 for every edge (segment_sum of ones over src)
// ---------------------------------------------------------------------------
__global__ void cheb_deg_kernel(const long long* __restrict__ src,
                                float* __restrict__ deg) {
    int e = blockIdx.x * blockDim.x + threadIdx.x;
    if (e < N_EDGES) {
        long long s = src[e];
        unsafeAtomicAdd(&deg[s], 1.0f);   // global_atomic_add_f32
    }
}

// ---------------------------------------------------------------------------
// deg -> deg^{-1/2} in place (counts are >=0 integers, so max(deg,1)=deg
// whenever deg>0)
// ---------------------------------------------------------------------------
__global__ void cheb_dis_kernel(float* __restrict__ deg) {
    int i = blockIdx.x * blockDim.x + threadIdx.x;
    if (i < N_NODES) {
        float d = deg[i];
        deg[i] = (d > 0.0f) ? rsqrtf(d) : 0.0f;
    }
}

// ---------------------------------------------------------------------------
// hout[dst] += (-dis[src]*dis[dst]) * hin[src]   (scatter-add, 8 lanes/edge,
// float4 per lane -> 51.2M f32 atomics, all L2-resident)
// ---------------------------------------------------------------------------
__global__ void cheb_prop_kernel(const float* __restrict__ hin,
                                 const long long* __restrict__ src,
                                 const long long* __restrict__ dst,
                                 const float* __restrict__ dis,
                                 float* __restrict__ hout) {
    unsigned tid = blockIdx.x * blockDim.x + threadIdx.x;
    unsigned e = tid >> 3;
    unsigned j = (tid & 7u) * 4u;
    if (e < N_EDGES) {
        long long s = src[e];
        long long d = dst[e];
        float w = -dis[s] * dis[d];
        const float4 hv = *(const float4*)(hin + (size_t)s * IN_F + j);
        float* o = hout + (size_t)d * IN_F + j;
        unsafeAtomicAdd(o + 0, w * hv.x);
        unsafeAtomicAdd(o + 1, w * hv.y);
        unsafeAtomicAdd(o + 2, w * hv.z);
        unsafeAtomicAdd(o + 3, w * hv.w);
    }
}

// ---------------------------------------------------------------------------
// out = relu( x@(W0-W2) + tx1@W1 + p2@(2*W2) + b )
// One wave per 16-row tile; all 24 B operands hoisted to registers, then a
// clean stream of 24x { global_load_b64 (A) ; v_wmma_f32_16x16x4_f32 }.
// ---------------------------------------------------------------------------
__global__ void __launch_bounds__(256)
cheb_gemm_kernel(const float* __restrict__ x,
                 const float* __restrict__ tx1,
                 const float* __restrict__ p2,
                 const float* __restrict__ W,
                 const float* __restrict__ bias,
                 float* __restrict__ out) {
    const int wave = threadIdx.x >> 5;
    const int tile = blockIdx.x * (blockDim.x >> 5) + wave;
    if (tile >= NTILES) return;                 // wave-uniform: EXEC stays all-1

    const int lane  = threadIdx.x & 31;
    const int laneM = lane & 15;                // A: row, B/C/D: column N
    const int hi    = lane >> 4;                // lane half: K pair sel / M+8
    const int m0    = tile * 16;

    // -------- Hoist all B operands (tile-invariant, depend only on lane) ----
    // bv[c*8+s] = B values for chain c, K-step s:
    //   component i holds B[4*s + 2*hi + i][laneM], zero for pad columns N>=8
    v2f bv[24];
    #pragma unroll
    for (int i = 0; i < 24; ++i) { bv[i].x = 0.0f; bv[i].y = 0.0f; }
    if (laneM < OUT_F) {
        #pragma unroll
        for (int s = 0; s < 8; ++s) {
            const int k0 = 4 * s + 2 * hi;
            const float* w0 = W + (0 * IN_F + k0) * OUT_F + laneM;
            const float* w1 = W + (1 * IN_F + k0) * OUT_F + laneM;
            const float* w2 = W + (2 * IN_F + k0) * OUT_F + laneM;
            bv[0 * 8 + s].x = w0[0]     - w2[0];        // W0 - W2
            bv[0 * 8 + s].y = w0[OUT_F] - w2[OUT_F];
            bv[1 * 8 + s].x = w1[0];                    // W1
            bv[1 * 8 + s].y = w1[OUT_F];
            bv[2 * 8 + s].x = 2.0f * w2[0];             // 2*W2
            bv[2 * 8 + s].y = 2.0f * w2[OUT_F];
        }
    }

    // -------- 3 accumulation chains, 8 x V_WMMA_F32_16X16X4_F32 each --------
    v8f acc = {};
    #pragma unroll
    for (int c = 0; c < 3; ++c) {
        const float* A = (c == 0) ? x : (c == 1) ? tx1 : p2;
        const float* arow = A + (size_t)(m0 + laneM) * IN_F + 2 * hi;
        #pragma unroll
        for (int s = 0; s < 8; ++s) {
            // A operand: lane holds A[m0+laneM][4*s+2*hi .. +1]
            v2f a = *(const v2f*)(arow + 4 * s);
            acc = __builtin_amdgcn_wmma_f32_16x16x4_f32(
                /*neg_a=*/false, a, /*neg_b=*/false, bv[c * 8 + s],
                /*c_mod=*/(short)0, acc, /*reuse_a=*/false, /*reuse_b=*/false);
        }
    }

    // C/D layout: VGPR v -> row m0+v (lanes 0-15) / m0+v+8 (lanes 16-31),
    // lane -> column N. Only N<8 is real output.
    if (laneM < OUT_F) {
        const float bvs = bias[laneM];
        #pragma unroll
        for (int v = 0; v < 8; ++v) {
            const int row = m0 + v + 8 * hi;
            float val = acc[v] + bvs;
            out[(size_t)row * OUT_F + laneM] = fmaxf(val, 0.0f);
        }
    }
}

// ---------------------------------------------------------------------------
extern "C" void kernel_launch(void* const* d_in, const int* in_sizes, int n_in,
                              void* d_out, int out_size, void* d_ws, size_t ws_size,
                              hipStream_t stream) {
    const float*     x  = (const float*)d_in[0];
    const long long* ei = (const long long*)d_in[1];   // int64 per reference
    const float*     W  = (const float*)d_in[2];
    const float*     b  = (const float*)d_in[3];
    float*           out = (float*)d_out;

    const long long* src = ei;              // edge_index[0]
    const long long* dst = ei + N_EDGES;    // edge_index[1]

    // Workspace layout: deg/dis [N] | tx1 [N*32] | p2 [N*32]  (~26 MB)
    float* deg = (float*)d_ws;
    float* tx1 = deg + N_NODES;
    float* p2  = tx1 + (size_t)N_NODES * IN_F;

    const int zn = N_NODES * (1 + 2 * IN_F);
    cheb_zero_kernel<<<(zn + 255) / 256, 256, 0, stream>>>((float*)d_ws, zn);

    cheb_deg_kernel<<<(N_EDGES + 255) / 256, 256, 0, stream>>>(src, deg);
    cheb_dis_kernel<<<(N_NODES + 255) / 256, 256, 0, stream>>>(deg);

    const int prop_threads = N_EDGES * 8;
    cheb_prop_kernel<<<(prop_threads + 255) / 256, 256, 0, stream>>>(x,   src, dst, deg, tx1);
    cheb_prop_kernel<<<(prop_threads + 255) / 256, 256, 0, stream>>>(tx1, src, dst, deg, p2);

    cheb_gemm_kernel<<<(NTILES + 7) / 8, 256, 0, stream>>>(x, tx1, p2, W, b, out);
}